// Rasterize_3582002725542
// MI455X (gfx1250) — compile-verified
//
#include <hip/hip_runtime.h>
#include <math.h>

typedef __attribute__((ext_vector_type(2))) float v2f;
typedef __attribute__((ext_vector_type(8))) float v8f;

#define IMG_H 256
#define IMG_W 256
#define NBATCH 4
#define NVERT 1026
#define NTRI 2048

// ---------------------------------------------------------------------------
// Kernel 0: build M = Perspective @ LookAt from eye/center/up (single thread)
// ---------------------------------------------------------------------------
__global__ void build_matrix_kernel(const float* __restrict__ eye,
                                    const float* __restrict__ center,
                                    const float* __restrict__ up,
                                    float* __restrict__ Mw) {
    if (threadIdx.x != 0 || blockIdx.x != 0) return;
    float ex = eye[0], ey = eye[1], ez = eye[2];
    float fx = center[0] - ex, fy = center[1] - ey, fz = center[2] - ez;
    float fl = 1.0f / sqrtf(fx * fx + fy * fy + fz * fz);
    fx *= fl; fy *= fl; fz *= fl;
    float ux = up[0], uy = up[1], uz = up[2];
    float sx = fy * uz - fz * uy;
    float sy = fz * ux - fx * uz;
    float sz = fx * uy - fy * ux;
    float sl = 1.0f / sqrtf(sx * sx + sy * sy + sz * sz);
    sx *= sl; sy *= sl; sz *= sl;
    float wx = sy * fz - sz * fy;     // u = cross(s, f)
    float wy = sz * fx - sx * fz;
    float wz = sx * fy - sy * fx;
    // View matrix rows: [s, t0] [u, t1] [-f, t2] [0 0 0 1], t = -R@eye
    float V[4][4] = {
        { sx,  sy,  sz, -(sx * ex + sy * ey + sz * ez) },
        { wx,  wy,  wz, -(wx * ex + wy * ey + wz * ez) },
        { -fx, -fy, -fz,  (fx * ex + fy * ey + fz * ez) },
        { 0.f, 0.f, 0.f, 1.f }
    };
    const float nearp = 0.01f, farp = 10.0f;
    float focal = 1.0f / tanf(40.0f * 3.14159265358979323846f / 360.0f);
    float pa = (farp + nearp) / (nearp - farp);
    float pb = 2.0f * farp * nearp / (nearp - farp);
    float P[4][4] = {
        { focal, 0.f,   0.f, 0.f },   // aspect = W/H = 1
        { 0.f,   focal, 0.f, 0.f },
        { 0.f,   0.f,   pa,  pb  },
        { 0.f,   0.f,  -1.f, 0.f }
    };
    for (int i = 0; i < 4; ++i)
        for (int j = 0; j < 4; ++j) {
            float acc = 0.f;
            for (int k = 0; k < 4; ++k) acc += P[i][k] * V[k][j];
            Mw[i * 4 + j] = acc;   // row-major M[i][k]
        }
}

// ---------------------------------------------------------------------------
// Kernel 1: vertex transform via V_WMMA_F32_16X16X4_F32.
// One wave (32 threads) transforms 16 vertices:
//   A (16x4): row v = [x,y,z,1] of vertex v  -> D(16x16)[v][i] = clip_i(v)
//   B (4x16): B[k][n] = M[n][k] (only n<4 columns meaningful)
// A layout (ISA 7.12.2): lanes 0-15 hold K=0(v0),K=1(v1); lanes 16-31 K=2,K=3.
// D layout: VGPR j, lanes 0-15 -> row j, lanes 16-31 -> row j+8; col = lane&15.
// ---------------------------------------------------------------------------
__global__ __launch_bounds__(32) void transform_wmma_kernel(
        const float* __restrict__ verts, const float* __restrict__ Mw,
        float* __restrict__ ndc, int total) {
    __shared__ float lds[16 * 16];
    const int lane = threadIdx.x;
    const int half = lane >> 4;       // 0: K=0,1   1: K=2,3
    const int l    = lane & 15;
    const int g    = blockIdx.x * 16 + l;
    const int gi   = (g < total) ? g : (total - 1);   // clamp (no divergence)

    const float* vp = verts + (size_t)gi * 3;
    v2f a;
    a.x = half ? vp[2] : vp[0];
    a.y = half ? 1.0f  : vp[1];

    const int k0 = half ? 2 : 0;
    v2f bmat;
    bmat.x = (l < 4) ? Mw[l * 4 + k0]     : 0.0f;
    bmat.y = (l < 4) ? Mw[l * 4 + k0 + 1] : 0.0f;

    v8f c = {};
    v8f d = __builtin_amdgcn_wmma_f32_16x16x4_f32(
        /*neg_a=*/false, a, /*neg_b=*/false, bmat,
        /*c_mod=*/(short)0, c, /*reuse_a=*/false, /*reuse_b=*/false);

    // Scatter D to LDS as clip[vertex][component]
#pragma unroll
    for (int j = 0; j < 8; ++j)
        lds[(j + half * 8) * 16 + l] = d[j];
    __syncthreads();

    if (half == 0 && g < total) {
        float x = lds[l * 16 + 0];
        float y = lds[l * 16 + 1];
        float z = lds[l * 16 + 2];
        float w = lds[l * 16 + 3];
        float iw = 1.0f / w;
        ndc[g * 3 + 0] = x * iw;
        ndc[g * 3 + 1] = y * iw;
        ndc[g * 3 + 2] = z * iw;
    }
}

// ---------------------------------------------------------------------------
// Kernel 2: rasterize + shade. 16x16 pixel tile per block (256 threads),
// triangles streamed through LDS in chunks of 256 (16 floats of setup each).
// ---------------------------------------------------------------------------
__global__ __launch_bounds__(256) void raster_kernel(
        const float* __restrict__ ndc, const int* __restrict__ faces,
        const float* __restrict__ verts, const float* __restrict__ normals,
        float* __restrict__ outPos, float* __restrict__ outNrm,
        int* __restrict__ outVid, float* __restrict__ outBary) {
    __shared__ float lds[256 * 16];

    const int b   = blockIdx.z;
    const int x   = blockIdx.x * 16 + threadIdx.x;
    const int y   = blockIdx.y * 16 + threadIdx.y;
    const int tid = threadIdx.y * 16 + threadIdx.x;

    const float px = ((float)x + 0.5f) * (2.0f / (float)IMG_W) - 1.0f;
    const float py = ((float)y + 0.5f) * (2.0f / (float)IMG_H) - 1.0f;

    const float* nb = ndc + (size_t)b * NVERT * 3;

    float zbuf = 1e30f;
    int   best = -1;
    float bb0 = 0.f, bb1 = 0.f, bb2 = 0.f;

    for (int chunk = 0; chunk < NTRI; chunk += 256) {
        __syncthreads();
        {   // cooperative triangle setup into LDS: one triangle per thread
            int t  = chunk + tid;
            int i0 = faces[t * 3 + 0];
            int i1 = faces[t * 3 + 1];
            int i2 = faces[t * 3 + 2];
            float x0 = nb[i0 * 3 + 0], y0 = nb[i0 * 3 + 1], z0 = nb[i0 * 3 + 2];
            float x1 = nb[i1 * 3 + 0], y1 = nb[i1 * 3 + 1], z1 = nb[i1 * 3 + 2];
            float x2 = nb[i2 * 3 + 0], y2 = nb[i2 * 3 + 1], z2 = nb[i2 * 3 + 2];
            float e1x = x1 - x0, e1y = y1 - y0;
            float e2x = x2 - x0, e2y = y2 - y0;
            float det = e1x * e2y - e2x * e1y;
            float inv = 1.0f / ((det == 0.0f) ? 1.0f : det);
            float valid = (det != 0.0f) ? 1.0f : 0.0f;
            const float mg = 2.0f / (float)IMG_W;   // one-pixel safety margin
            float mnx = fminf(x0, fminf(x1, x2)) - mg;
            float mxx = fmaxf(x0, fmaxf(x1, x2)) + mg;
            float mny = fminf(y0, fminf(y1, y2)) - mg;
            float mxy = fmaxf(y0, fmaxf(y1, y2)) + mg;
            float* s = &lds[tid * 16];
            s[0] = x0;  s[1] = y0;  s[2] = e1x; s[3] = e1y;
            s[4] = e2x; s[5] = e2y; s[6] = inv; s[7] = z0;
            s[8] = z1;  s[9] = z2;  s[10] = valid; s[11] = 0.f;
            s[12] = mnx; s[13] = mxx; s[14] = mny; s[15] = mxy;
            if (chunk + 256 < NTRI)   // hint next chunk's face data into cache
                __builtin_prefetch(&faces[(chunk + 256 + tid) * 3], 0, 1);
        }
        __syncthreads();

        for (int t2 = 0; t2 < 256; ++t2) {
            const float* s = &lds[t2 * 16];     // LDS broadcast (uniform addr)
            if (px < s[12] || px > s[13] || py < s[14] || py > s[15]) continue;
            float dx = px - s[0], dy = py - s[1];
            float c1 = (dx * s[5] - dy * s[4]) * s[6];
            float c2 = (s[2] * dy - s[3] * dx) * s[6];
            float c0 = 1.0f - c1 - c2;
            float z  = c0 * s[7] + c1 * s[8] + c2 * s[9];
            bool hit = (s[10] != 0.0f) && (c0 >= 0.0f) && (c1 >= 0.0f) &&
                       (c2 >= 0.0f) && (z >= -1.0f) && (z <= 1.0f) && (z < zbuf);
            if (hit) {
                zbuf = z; best = chunk + t2;
                bb0 = c0; bb1 = c1; bb2 = c2;
            }
        }
    }

    // ------------------- shading epilogue -------------------
    int idx = (best >= 0) ? best : 0;                 // clip(tid, 0)
    int v0 = faces[idx * 3 + 0];
    int v1 = faces[idx * 3 + 1];
    int v2 = faces[idx * 3 + 2];
    const float* vb = verts   + (size_t)b * NVERT * 3;
    const float* rb = normals + (size_t)b * NVERT * 3;

    float pxp = bb0 * vb[v0 * 3 + 0] + bb1 * vb[v1 * 3 + 0] + bb2 * vb[v2 * 3 + 0];
    float pyp = bb0 * vb[v0 * 3 + 1] + bb1 * vb[v1 * 3 + 1] + bb2 * vb[v2 * 3 + 1];
    float pzp = bb0 * vb[v0 * 3 + 2] + bb1 * vb[v1 * 3 + 2] + bb2 * vb[v2 * 3 + 2];

    float nxp = bb0 * rb[v0 * 3 + 0] + bb1 * rb[v1 * 3 + 0] + bb2 * rb[v2 * 3 + 0];
    float nyp = bb0 * rb[v0 * 3 + 1] + bb1 * rb[v1 * 3 + 1] + bb2 * rb[v2 * 3 + 1];
    float nzp = bb0 * rb[v0 * 3 + 2] + bb1 * rb[v1 * 3 + 2] + bb2 * rb[v2 * 3 + 2];
    float rn  = 1.0f / (sqrtf(nxp * nxp + nyp * nyp + nzp * nzp) + 1e-12f);

    size_t p = (((size_t)b * IMG_H + y) * IMG_W + x) * 3;
    outPos[p + 0] = pxp;      outPos[p + 1] = pyp;      outPos[p + 2] = pzp;
    outNrm[p + 0] = nxp * rn; outNrm[p + 1] = nyp * rn; outNrm[p + 2] = nzp * rn;
    outVid[p + 0] = (best >= 0) ? v0 : -1;
    outVid[p + 1] = (best >= 0) ? v1 : -1;
    outVid[p + 2] = (best >= 0) ? v2 : -1;
    outBary[p + 0] = bb0; outBary[p + 1] = bb1; outBary[p + 2] = bb2;
}

// ---------------------------------------------------------------------------
extern "C" void kernel_launch(void* const* d_in, const int* in_sizes, int n_in,
                              void* d_out, int out_size, void* d_ws, size_t ws_size,
                              hipStream_t stream) {
    const float* vertices = (const float*)d_in[0];   // [B,V,3]
    const int*   faces    = (const int*)  d_in[1];   // [T,3]
    const float* normals  = (const float*)d_in[2];   // [B,V,3]
    // d_in[3] colors (unused by returned buffers)
    const float* eye      = (const float*)d_in[4];
    const float* center   = (const float*)d_in[5];
    const float* world_up = (const float*)d_in[6];
    // d_in[7], d_in[8] lights (unused)

    float* Mw  = (float*)d_ws;        // 16 floats
    float* ndc = Mw + 16;             // B*V*3 floats (~48 KB)
    const int total = NBATCH * NVERT;

    build_matrix_kernel<<<1, 1, 0, stream>>>(eye, center, world_up, Mw);

    const int tiles = (total + 15) / 16;   // 257 waves, one WMMA each
    transform_wmma_kernel<<<tiles, 32, 0, stream>>>(vertices, Mw, ndc, total);

    const size_t plane = (size_t)NBATCH * IMG_H * IMG_W * 3;  // 786432
    float* outPos  = (float*)d_out;
    float* outNrm  = outPos + plane;
    int*   outVid  = (int*)d_out + 2 * plane;
    float* outBary = (float*)d_out + 3 * plane;

    dim3 grid(IMG_W / 16, IMG_H / 16, NBATCH);
    dim3 block(16, 16, 1);
    raster_kernel<<<grid, block, 0, stream>>>(ndc, faces, vertices, normals,
                                              outPos, outNrm, outVid, outBary);
}